// LargeMarginModule_CosineLoss_34454227649153
// MI455X (gfx1250) — compile-verified
//
#include <hip/hip_runtime.h>
#include <math.h>

typedef __attribute__((ext_vector_type(2))) float v2f;
typedef __attribute__((ext_vector_type(8))) float v8f;

#define B_ROWS 2048
#define D_DIM  512
#define C_COLS 32000

#define BM 64
#define BN 128
#define BK 32
#define LDA 36   // padded LDS stride (floats): 16B-aligned rows, conflict-free b64 reads
#define LDB 36
#define NCHUNK (D_DIM / BK)

// ---------------------------------------------------------------------------
// CDNA5 async global->LDS copy (GV mode), tracked by ASYNCcnt.
//   global_load_async_to_lds_b128  vdst(LDS byte addr), v[addr64], off
// Generic pointers to __shared__ keep the LDS byte offset in addr[31:0].
// ---------------------------------------------------------------------------
__device__ __forceinline__ void async_copy_b128(void* lds_dst, const void* gsrc) {
  uint32_t ldsa = (uint32_t)(uintptr_t)lds_dst;
  uint64_t ga   = (uint64_t)(uintptr_t)gsrc;
  asm volatile("global_load_async_to_lds_b128 %0, %1, off"
               :: "v"(ldsa), "v"(ga) : "memory");
}
__device__ __forceinline__ void wait_async_all() {
  asm volatile("s_wait_asynccnt 0x0" ::: "memory");
}

// ---------------------------------------------------------------------------
// Row L2-norm reciprocal: one wave (32 lanes) per row of 512 floats.
// inv[row] = 1 / max(sqrt(sum(v^2)), 1e-12)   (matches F.normalize semantics)
// ---------------------------------------------------------------------------
__global__ __launch_bounds__(256) void rownorm_inv_kernel(
    const float* __restrict__ src, float* __restrict__ inv, int rows) {
  const int lane = threadIdx.x & 31;
  const int wave = (blockIdx.x * (blockDim.x >> 5)) + (threadIdx.x >> 5);
  if (wave >= rows) return;

  const float4* p = (const float4*)(src + (size_t)wave * D_DIM);
  float s = 0.0f;
#pragma unroll
  for (int i = 0; i < 4; ++i) {              // 512 floats = 128 float4, 4/lane
    float4 v = p[lane + 32 * i];
    s = fmaf(v.x, v.x, s);
    s = fmaf(v.y, v.y, s);
    s = fmaf(v.z, v.z, s);
    s = fmaf(v.w, v.w, s);
  }
#pragma unroll
  for (int off = 16; off > 0; off >>= 1) s += __shfl_xor(s, off, 32);
  if (lane == 0) inv[wave] = 1.0f / fmaxf(sqrtf(s), 1e-12f);
}

// ---------------------------------------------------------------------------
// Fused GEMM + scale + margin-scatter: V_WMMA_F32_16X16X4_F32, double-buffered
// LDS fed by GLOBAL_LOAD_ASYNC_TO_LDS_B128, NT epilogue stores.
// Workgroup: 256 threads (8 waves), 64x128 tile; wave -> 16x64 strip.
// ---------------------------------------------------------------------------
__global__ __launch_bounds__(256) void lmcos_gemm_kernel(
    const float* __restrict__ x, const float* __restrict__ w,
    const int* __restrict__ targets, const float* __restrict__ invx,
    const float* __restrict__ invw, float* __restrict__ out) {
  __shared__ alignas(16) float As[2][BM * LDA];   // 2 x 64x(32+4)
  __shared__ alignas(16) float Bs[2][BN * LDB];   // 2 x 128x(32+4)

  const int tid   = threadIdx.x;
  const int lane  = tid & 31;
  const int wv    = tid >> 5;      // 0..7
  const int waveM = wv & 3;        // 0..3  -> 16-row strip within 64
  const int waveN = wv >> 2;       // 0..1  -> 64-col strip within 128
  const int mtile = blockIdx.y * BM;
  const int ntile = blockIdx.x * BN;

  // Per-thread staging slots: A = 64x32 floats (512 float4, 2/thread),
  //                           B = 128x32 floats (1024 float4, 4/thread).
  int arow_ld[2], acol_ld[2], brow_ld[4], bcol_ld[4];
#pragma unroll
  for (int i = 0; i < 2; ++i) {
    int idx = tid + 256 * i;
    arow_ld[i] = idx >> 3;
    acol_ld[i] = (idx & 7) << 2;
  }
#pragma unroll
  for (int i = 0; i < 4; ++i) {
    int idx = tid + 256 * i;
    brow_ld[i] = idx >> 3;
    bcol_ld[i] = (idx & 7) << 2;
  }

  v8f acc[4];
#pragma unroll
  for (int j = 0; j < 4; ++j) acc[j] = (v8f){0, 0, 0, 0, 0, 0, 0, 0};

  // WMMA fp32 16x16x4 fragment addressing: lanes 0..15 hold K0/K1 for row
  // M=lane, lanes 16..31 hold K2/K3 for row M=lane-16 (same for B columns).
  const int frow = lane & 15;
  const int koff = (lane >> 4) << 1;
  const int arow = waveM * 16 + frow;       // A row within LDS tile
  const int brow = waveN * 64 + frow;       // B base row (output column)

  // ---- preload chunk 0 into buffer 0 ----
#pragma unroll
  for (int i = 0; i < 2; ++i)
    async_copy_b128(&As[0][arow_ld[i] * LDA + acol_ld[i]],
                    x + (size_t)(mtile + arow_ld[i]) * D_DIM + acol_ld[i]);
#pragma unroll
  for (int i = 0; i < 4; ++i)
    async_copy_b128(&Bs[0][brow_ld[i] * LDB + bcol_ld[i]],
                    w + (size_t)(ntile + brow_ld[i]) * D_DIM + bcol_ld[i]);

  for (int it = 0; it < NCHUNK; ++it) {
    const int cur = it & 1;
    wait_async_all();     // my wave's chunk `it` is in LDS (in-order completion)
    __syncthreads();      // all waves' chunk landed; other buffer free for reuse

    if (it + 1 < NCHUNK) {
      const int k1 = (it + 1) * BK;
#pragma unroll
      for (int i = 0; i < 2; ++i)
        async_copy_b128(&As[1 - cur][arow_ld[i] * LDA + acol_ld[i]],
                        x + (size_t)(mtile + arow_ld[i]) * D_DIM + k1 + acol_ld[i]);
#pragma unroll
      for (int i = 0; i < 4; ++i)
        async_copy_b128(&Bs[1 - cur][brow_ld[i] * LDB + bcol_ld[i]],
                        w + (size_t)(ntile + brow_ld[i]) * D_DIM + k1 + bcol_ld[i]);
    }

    const float* __restrict__ Ash = As[cur];
    const float* __restrict__ Bsh = Bs[cur];
#pragma unroll
    for (int kk = 0; kk < BK; kk += 4) {
      v2f a;
      a.x = Ash[arow * LDA + kk + koff];
      a.y = Ash[arow * LDA + kk + koff + 1];
#pragma unroll
      for (int j = 0; j < 4; ++j) {
        v2f b;
        b.x = Bsh[(brow + j * 16) * LDB + kk + koff];
        b.y = Bsh[(brow + j * 16) * LDB + kk + koff + 1];
        acc[j] = __builtin_amdgcn_wmma_f32_16x16x4_f32(
            /*neg_a=*/false, a, /*neg_b=*/false, b,
            /*c_mod=*/(short)0, acc[j], /*reuse_a=*/false, /*reuse_b=*/false);
      }
    }
  }

  // ---- epilogue: scale + margin, streaming NT stores of both halves ----
  // C/D layout: VGPR i, lanes 0..15 -> M=i, lanes 16..31 -> M=i+8; N = lane&15.
  const int mrow0 = mtile + waveM * 16 + ((lane >> 4) << 3);
  float sx[8];
  int   tg[8];
#pragma unroll
  for (int i = 0; i < 8; ++i) {
    int m = mrow0 + i;
    sx[i] = 32.0f * invx[m];   // SCALE folded with x-row inverse norm
    tg[i] = targets[m];
  }

  float* __restrict__ loss_out = out;                                // tuple[0]
  float* __restrict__ pred_out = out + (size_t)B_ROWS * C_COLS;      // tuple[1]

#pragma unroll
  for (int j = 0; j < 4; ++j) {
    const int c = ntile + waveN * 64 + j * 16 + (lane & 15);
    const float sw = invw[c];
#pragma unroll
    for (int i = 0; i < 8; ++i) {
      const float val = acc[j][i] * sx[i] * sw;
      const size_t off = (size_t)(mrow0 + i) * C_COLS + c;
      __builtin_nontemporal_store(val, &pred_out[off]);
      __builtin_nontemporal_store(val - ((c == tg[i]) ? 16.0f : 0.0f),
                                  &loss_out[off]);   // -M*SCALE = -16
    }
  }
}

// ---------------------------------------------------------------------------
extern "C" void kernel_launch(void* const* d_in, const int* in_sizes, int n_in,
                              void* d_out, int out_size, void* d_ws, size_t ws_size,
                              hipStream_t stream) {
  const float* x   = (const float*)d_in[0];   // [2048, 512]
  const float* w   = (const float*)d_in[1];   // [32000, 512]
  const int*   tgt = (const int*)d_in[2];     // [2048]
  float*       out = (float*)d_out;           // [2*2048*32000]

  float* invx = (float*)d_ws;                 // 2048 floats
  float* invw = invx + B_ROWS;                // 32000 floats

  rownorm_inv_kernel<<<B_ROWS / 8, 256, 0, stream>>>(x, invx, B_ROWS);
  rownorm_inv_kernel<<<C_COLS / 8, 256, 0, stream>>>(w, invw, C_COLS);

  dim3 grid(C_COLS / BN, B_ROWS / BM);        // 250 x 32 workgroups
  lmcos_gemm_kernel<<<grid, 256, 0, stream>>>(x, w, tgt, invx, invw, out);
}